// EpiGNN_3951369912905
// MI455X (gfx1250) — compile-verified
//
#include <hip/hip_runtime.h>
#include <math.h>

typedef __attribute__((ext_vector_type(2))) float v2f;
typedef __attribute__((ext_vector_type(8))) float v8f;

constexpr int BSZ = 4, TIN = 16, MNODE = 2000, NFEATC = 4;
constexpr int HIDA = 32, HIDR = 40, TOUTC = 12;
constexpr int HEADS = 4, DHEAD = 10, NLAYER = 3;
constexpr int NODES = BSZ * MNODE;      // 8000
constexpr int MASKW = 64;               // u32 mask words per adjacency row
constexpr float INV_SQRT_BN = 0.99999500003749973f; // 1/sqrt(1+1e-5)

// ---------------------------------------------------------------- conv stage
struct ConvP {
    const float* w[5];
    const float* b[5];
    const float* g[5];
    const float* e[5];
};

template<int KS, int DIL>
__device__ inline void conv_branch(const float xv[TIN][NFEATC],
                                   const float* __restrict__ w,
                                   const float* __restrict__ bb,
                                   const float* __restrict__ ga,
                                   const float* __restrict__ be,
                                   float* __restrict__ outc)
{
    constexpr int TP = TIN - (KS - 1) * DIL;   // valid conv length
    #pragma unroll
    for (int kk = 0; kk < 8; ++kk) {
        float gk = ga[kk] * INV_SQRT_BN, bek = be[kk], bk = bb[kk];
        float mx = -INFINITY;
        #pragma unroll
        for (int t = 0; t < TP; ++t) {
            float s = bk;
            #pragma unroll
            for (int dt = 0; dt < KS; ++dt)
                #pragma unroll
                for (int f = 0; f < NFEATC; ++f)
                    s += w[(kk * NFEATC + f) * KS + dt] * xv[t + dt * DIL][f];
            s = s * gk + bek;
            mx = fmaxf(mx, s);                 // maxpool over time (TP==1 -> identity)
        }
        outc[kk] = tanhf(mx);
    }
}

__global__ void conv_temp_kernel(const float* __restrict__ X, ConvP cp,
                                 float* __restrict__ temp)
{
    int idx = blockIdx.x * blockDim.x + threadIdx.x;
    if (idx >= NODES) return;
    int b = idx / MNODE, m = idx - b * MNODE;

    float xv[TIN][NFEATC];
    const float4* xp = (const float4*)X;       // X[b][t][m][0..3] is 16B aligned
    #pragma unroll
    for (int t = 0; t < TIN; ++t) {
        float4 v = xp[(b * TIN + t) * MNODE + m];
        xv[t][0] = v.x; xv[t][1] = v.y; xv[t][2] = v.z; xv[t][3] = v.w;
    }

    float oc[8];
    float* tr = temp + (size_t)idx * HIDR;
    conv_branch<3, 1>(xv, cp.w[0], cp.b[0], cp.g[0], cp.e[0], oc);
    #pragma unroll
    for (int kk = 0; kk < 8; ++kk) tr[kk] = tanhf(oc[kk]);          // double tanh
    conv_branch<5, 1>(xv, cp.w[1], cp.b[1], cp.g[1], cp.e[1], oc);
    #pragma unroll
    for (int kk = 0; kk < 8; ++kk) tr[8 + kk] = tanhf(oc[kk]);
    conv_branch<3, 2>(xv, cp.w[2], cp.b[2], cp.g[2], cp.e[2], oc);
    #pragma unroll
    for (int kk = 0; kk < 8; ++kk) tr[16 + kk] = tanhf(oc[kk]);
    conv_branch<5, 2>(xv, cp.w[3], cp.b[3], cp.g[3], cp.e[3], oc);
    #pragma unroll
    for (int kk = 0; kk < 8; ++kk) tr[24 + kk] = tanhf(oc[kk]);
    conv_branch<16, 1>(xv, cp.w[4], cp.b[4], cp.g[4], cp.e[4], oc);
    #pragma unroll
    for (int kk = 0; kk < 8; ++kk) tr[32 + kk] = tanhf(oc[kk]);
}

// ----------------------------------------------------------------- q/k stage
__global__ void qk_kernel(const float* __restrict__ temp,
                          const float* __restrict__ WQ, const float* __restrict__ bQ,
                          const float* __restrict__ WK, const float* __restrict__ bK,
                          float* __restrict__ q, float* __restrict__ k)
{
    int idx = blockIdx.x * blockDim.x + threadIdx.x;
    if (idx >= NODES) return;
    float t[HIDR];
    #pragma unroll
    for (int c = 0; c < HIDR; ++c) t[c] = temp[(size_t)idx * HIDR + c];
    for (int a = 0; a < HIDA; ++a) {
        float qa = bQ[a], ka = bK[a];
        #pragma unroll
        for (int c = 0; c < HIDR; ++c) {
            qa += WQ[a * HIDR + c] * t[c];
            ka += WK[a * HIDR + c] * t[c];
        }
        q[(size_t)idx * HIDA + a] = qa;
        k[(size_t)idx * HIDA + a] = ka;
    }
}

// Gram matrix G = K^T K (32x32 per batch) and Ksum = sum_j k_j.
__global__ void gram_kernel(const float* __restrict__ k,
                            float* __restrict__ G, float* __restrict__ Ksum)
{
    __shared__ float kb[32][33];
    int b = blockIdx.x;
    int tid = threadIdx.x;          // 0..1023
    int row = tid >> 5, col = tid & 31;
    float acc = 0.f, sacc = 0.f;
    for (int base = 0; base < MNODE; base += 32) {
        int j = base + row;
        kb[row][col] = (j < MNODE) ? k[(size_t)(b * MNODE + j) * HIDA + col] : 0.f;
        __syncthreads();
        #pragma unroll
        for (int r = 0; r < 32; ++r) acc += kb[r][row] * kb[r][col];
        if (row == 0) {
            #pragma unroll
            for (int r = 0; r < 32; ++r) sacc += kb[r][col];
        }
        __syncthreads();
    }
    G[b * 1024 + row * 32 + col] = acc;
    if (row == 0) Ksum[b * 32 + col] = sacc;
}

// ------------------------------------------------ adjacency single-pass stage
__global__ void __launch_bounds__(32) adjrow_kernel(const float* __restrict__ adj,
                                                    float* __restrict__ dsum,
                                                    unsigned* __restrict__ maskw)
{
    int bi = blockIdx.x;                 // 0..NODES-1
    int l = threadIdx.x;                 // 0..31
    const float* row = adj + (size_t)bi * MNODE;
    unsigned* mrow = maskw + (size_t)bi * MASKW;
    float s = 0.f;
    for (int c = 0; c < 63; ++c) {
        int col = c * 32 + l;
        float a = (col < MNODE) ? row[col] : 0.f;
        bool p = (col < MNODE) && (a > 0.f);
        unsigned wbits = __builtin_amdgcn_ballot_w32(p);
        if (l == 0) mrow[c] = wbits;
        s += a;
    }
    if (l == 0) mrow[63] = 0u;
    #pragma unroll
    for (int off = 16; off > 0; off >>= 1) s += __shfl_down(s, off, 32);
    if (l == 0) dsum[bi] = s;
}

// ----------------------------------- collapsed attention + positional encoders
__global__ void feat_kernel(const float* __restrict__ temp, const float* __restrict__ q,
                            const float* __restrict__ G, const float* __restrict__ Ksum,
                            const float* __restrict__ dsum,
                            const float* __restrict__ Wt, const float* __restrict__ bt,
                            const float* __restrict__ Ws, const float* __restrict__ bs_,
                            float* __restrict__ feat)
{
    int idx = blockIdx.x * blockDim.x + threadIdx.x;
    if (idx >= NODES) return;
    int b = idx / MNODE;
    float qv[HIDA];
    #pragma unroll
    for (int a = 0; a < HIDA; ++a) qv[a] = q[(size_t)idx * HIDA + a];
    const float* Gb = G + b * 1024;
    const float* Kb = Ksum + b * 32;
    float s = 0.f, n2 = 0.f;
    for (int a = 0; a < HIDA; ++a) {
        float ga = 0.f;
        #pragma unroll
        for (int c = 0; c < HIDA; ++c) ga += Gb[a * 32 + c] * qv[c];
        n2 += qv[a] * ga;
        s  += qv[a] * Kb[a];
    }
    float nrm = sqrtf(fmaxf(n2, 0.f));
    float att = s / fmaxf(nrm, 1e-12f);
    float d = dsum[idx];
    #pragma unroll
    for (int c = 0; c < HIDR; ++c)
        feat[(size_t)idx * HIDR + c] =
            temp[(size_t)idx * HIDR + c] + att * Wt[c] + bt[c] + d * Ws[c] + bs_[c];
}

// ------------------------------------------------------------- GAT projection
__global__ void wh_kernel(const float* __restrict__ ns, const float* __restrict__ gatW,
                          const float* __restrict__ gata,
                          float* __restrict__ Wh, float* __restrict__ e1,
                          float* __restrict__ e2)
{
    int idx = blockIdx.x * blockDim.x + threadIdx.x;
    if (idx >= NODES) return;
    float x[HIDR], w[HIDR];
    #pragma unroll
    for (int c = 0; c < HIDR; ++c) x[c] = ns[(size_t)idx * HIDR + c];
    for (int o = 0; o < HIDR; ++o) {
        float acc = 0.f;
        #pragma unroll
        for (int c = 0; c < HIDR; ++c) acc += gatW[o * HIDR + c] * x[c];
        w[o] = acc;
        Wh[(size_t)idx * HIDR + o] = acc;
    }
    int b = idx / MNODE, m = idx - b * MNODE;
    #pragma unroll
    for (int h = 0; h < HEADS; ++h) {
        float a1 = 0.f, a2 = 0.f;
        #pragma unroll
        for (int d = 0; d < DHEAD; ++d) {
            a1 += w[h * DHEAD + d] * gata[h * 2 * DHEAD + d];
            a2 += w[h * DHEAD + d] * gata[h * 2 * DHEAD + DHEAD + d];
        }
        e1[(size_t)(b * HEADS + h) * MNODE + m] = a1;
        e2[(size_t)(b * HEADS + h) * MNODE + m] = a2;
    }
}

__global__ void e2max_kernel(const float* __restrict__ e2, float* __restrict__ e2max)
{
    __shared__ float sm[256];
    int bh = blockIdx.x;
    float mx = -INFINITY;
    for (int i = threadIdx.x; i < MNODE; i += 256)
        mx = fmaxf(mx, e2[(size_t)bh * MNODE + i]);
    sm[threadIdx.x] = mx;
    __syncthreads();
    for (int s = 128; s > 0; s >>= 1) {
        if (threadIdx.x < s) sm[threadIdx.x] = fmaxf(sm[threadIdx.x], sm[threadIdx.x + s]);
        __syncthreads();
    }
    if (threadIdx.x == 0) e2max[bh] = sm[0];
}

// -------------------- GAT masked softmax-aggregate via V_WMMA_F32_16X16X4_F32
// One wave per (b, h, 16-row tile). A[m][k] = softmax weight for (row m, col jc+k),
// B[k][n] = Wh[jc+k][head col n] padded with col15 = 1.0 so C[:,15] accumulates the
// softmax denominator in the same WMMA chain. Inner loop processes one 32-bit mask
// word (32 columns = 8 WMMAs) per trip; mask selection is a branchless multiply so
// EXEC stays all-ones and no saveexec branches appear in the hot loop.
__global__ void __launch_bounds__(32) gat_attn_kernel(
    const float* __restrict__ Wh, const float* __restrict__ e1,
    const float* __restrict__ e2, const float* __restrict__ e2max,
    const unsigned* __restrict__ maskw, float* __restrict__ nsout)
{
    int itile = blockIdx.x;             // 0..124
    int h = blockIdx.y;                 // 0..3
    int b = blockIdx.z;                 // 0..3
    int lane = threadIdx.x;
    int half = lane >> 4;               // which K pair this lane supplies
    int ln = lane & 15;
    int i0 = itile * 16;
    int bh = b * HEADS + h;

    const float* e1r = e1 + (size_t)bh * MNODE;
    const float* e2r = e2 + (size_t)bh * MNODE;
    float E2M = e2max[bh];

    int myrow = i0 + ln;                                 // A-matrix row for this lane
    float e1v = e1r[myrow];
    float tmax = e1v + E2M;
    float Mi = tmax > 0.f ? tmax : 0.2f * tmax;          // lrelu monotone => row max bound

    const unsigned* mrow = maskw + (size_t)(b * MNODE + myrow) * MASKW;
    const float* Whb = Wh + (size_t)b * MNODE * HIDR + h * DHEAD;

    // branchless B-column constants (keeps EXEC all-ones around the WMMA)
    int lnc = (ln < DHEAD) ? ln : 0;
    float fsel = (ln < DHEAD) ? 1.f : 0.f;
    float fden = (ln == 15) ? 1.f : 0.f;

    v8f c = {0.f, 0.f, 0.f, 0.f, 0.f, 0.f, 0.f, 0.f};
    int kb = half * 2;

    for (int jw = 0; jw < MNODE; jw += 32) {
        unsigned word = mrow[jw >> 5];                   // one mask word / 32 cols
        __builtin_prefetch(Whb + (size_t)(jw + 32) * HIDR, 0, 0); // global_prefetch_b8
        #pragma unroll
        for (int u = 0; u < 8; ++u) {
            int bit0 = u * 4 + kb;                       // lane's two mask bits
            int j0 = jw + bit0;
            float2 ev = *(const float2*)(e2r + j0);      // j0 even -> b64 load
            float sa = (float)((word >> bit0) & 1u);
            float sb = (float)((word >> (bit0 + 1)) & 1u);
            float va = e1v + ev.x;
            float vb = e1v + ev.y;
            va = va > 0.f ? va : 0.2f * va;              // leaky-relu
            vb = vb > 0.f ? vb : 0.2f * vb;
            float wa = sa * __expf(va - Mi);             // multiply-mask: no saveexec
            float wb = sb * __expf(vb - Mi);
            v2f A; A.x = wa; A.y = wb;

            float b0 = Whb[(size_t)j0 * HIDR + lnc] * fsel + fden;
            float b1 = Whb[(size_t)(j0 + 1) * HIDR + lnc] * fsel + fden;
            v2f B; B.x = b0; B.y = b1;

            c = __builtin_amdgcn_wmma_f32_16x16x4_f32(false, A, false, B,
                                                      (short)0, c, false, false);
        }
    }

    float* outb = nsout + (size_t)b * MNODE * HIDR + h * DHEAD;
    #pragma unroll
    for (int r = 0; r < 8; ++r) {
        float den = __shfl(c[r], half * 16 + 15, 32);    // C[:,15] holds denominator
        if (ln < DHEAD) {
            int m = r + half * 8;
            float val = c[r] / den;
            float res = val > 0.f ? val : expm1f(val);   // ELU
            outb[(size_t)(i0 + m) * HIDR + ln] = res;
        }
    }
}

// ----------------------------------------------------------------- output GEMM
__global__ void out_kernel(const float* __restrict__ ns0, const float* __restrict__ ns1,
                           const float* __restrict__ ns2, const float* __restrict__ feat,
                           const float* __restrict__ Wout, const float* __restrict__ bout,
                           float* __restrict__ out)
{
    int idx = blockIdx.x * blockDim.x + threadIdx.x;
    if (idx >= NODES) return;
    int b = idx / MNODE, i = idx - b * MNODE;
    float a0[HIDR], a1[HIDR], a2[HIDR], a3[HIDR];
    #pragma unroll
    for (int cidx = 0; cidx < HIDR; ++cidx) {
        a0[cidx] = ns0[(size_t)idx * HIDR + cidx];
        a1[cidx] = ns1[(size_t)idx * HIDR + cidx];
        a2[cidx] = ns2[(size_t)idx * HIDR + cidx];
        a3[cidx] = feat[(size_t)idx * HIDR + cidx];
    }
    for (int t = 0; t < TOUTC; ++t) {
        const float* wr = Wout + t * (4 * HIDR);
        float acc = bout[t];
        #pragma unroll
        for (int cidx = 0; cidx < HIDR; ++cidx) {
            acc += wr[cidx] * a0[cidx];
            acc += wr[HIDR + cidx] * a1[cidx];
            acc += wr[2 * HIDR + cidx] * a2[cidx];
            acc += wr[3 * HIDR + cidx] * a3[cidx];
        }
        out[((size_t)b * TOUTC + t) * MNODE + i] = acc;
    }
}

// =============================================================== host driver
extern "C" void kernel_launch(void* const* d_in, const int* in_sizes, int n_in,
                              void* d_out, int out_size, void* d_ws, size_t ws_size,
                              hipStream_t stream)
{
    (void)in_sizes; (void)n_in; (void)out_size; (void)ws_size;

    const float* X   = (const float*)d_in[0];
    const float* adj = (const float*)d_in[1];
    ConvP cp;
    for (int s = 0; s < 5; ++s) {
        cp.w[s] = (const float*)d_in[2 + 4 * s + 0];
        cp.b[s] = (const float*)d_in[2 + 4 * s + 1];
        cp.g[s] = (const float*)d_in[2 + 4 * s + 2];
        cp.e[s] = (const float*)d_in[2 + 4 * s + 3];
    }
    const float* WQ  = (const float*)d_in[22];
    const float* bQ  = (const float*)d_in[23];
    const float* WK  = (const float*)d_in[24];
    const float* bK  = (const float*)d_in[25];
    const float* Wt  = (const float*)d_in[26];
    const float* bt  = (const float*)d_in[27];
    const float* Ws  = (const float*)d_in[28];
    const float* bs_ = (const float*)d_in[29];
    const float* gatW[NLAYER] = { (const float*)d_in[30], (const float*)d_in[32],
                                  (const float*)d_in[34] };
    const float* gata[NLAYER] = { (const float*)d_in[31], (const float*)d_in[33],
                                  (const float*)d_in[35] };
    const float* Wout = (const float*)d_in[36];
    const float* bout = (const float*)d_in[37];
    float* out = (float*)d_out;

    // workspace carve-out (256B aligned slabs)
    size_t off = 0;
    char* wsb = (char*)d_ws;
    auto alloc = [&](size_t nbytes) -> void* {
        off = (off + 255) & ~(size_t)255;
        void* p = wsb + off;
        off += nbytes;
        return p;
    };
    float*    temp   = (float*)alloc((size_t)NODES * HIDR * 4);
    float*    q      = (float*)alloc((size_t)NODES * HIDA * 4);
    float*    k      = (float*)alloc((size_t)NODES * HIDA * 4);
    float*    G      = (float*)alloc((size_t)BSZ * 1024 * 4);
    float*    Ksum   = (float*)alloc((size_t)BSZ * 32 * 4);
    float*    dsum   = (float*)alloc((size_t)NODES * 4);
    unsigned* maskw  = (unsigned*)alloc((size_t)NODES * MASKW * 4);
    float*    feat   = (float*)alloc((size_t)NODES * HIDR * 4);
    float*    Wh     = (float*)alloc((size_t)NODES * HIDR * 4);
    float*    e1     = (float*)alloc((size_t)BSZ * HEADS * MNODE * 4);
    float*    e2     = (float*)alloc((size_t)BSZ * HEADS * MNODE * 4);
    float*    e2mx   = (float*)alloc((size_t)BSZ * HEADS * 4);
    float*    nsb[NLAYER];
    for (int l = 0; l < NLAYER; ++l)
        nsb[l] = (float*)alloc((size_t)NODES * HIDR * 4);

    const int TB = 128;
    const int NB = (NODES + TB - 1) / TB;   // 63

    conv_temp_kernel<<<NB, TB, 0, stream>>>(X, cp, temp);
    qk_kernel<<<NB, TB, 0, stream>>>(temp, WQ, bQ, WK, bK, q, k);
    gram_kernel<<<BSZ, 1024, 0, stream>>>(k, G, Ksum);
    adjrow_kernel<<<NODES, 32, 0, stream>>>(adj, dsum, maskw);
    feat_kernel<<<NB, TB, 0, stream>>>(temp, q, G, Ksum, dsum, Wt, bt, Ws, bs_, feat);

    const float* nsin = feat;
    for (int l = 0; l < NLAYER; ++l) {
        wh_kernel<<<NB, TB, 0, stream>>>(nsin, gatW[l], gata[l], Wh, e1, e2);
        e2max_kernel<<<BSZ * HEADS, 256, 0, stream>>>(e2, e2mx);
        gat_attn_kernel<<<dim3(MNODE / 16, HEADS, BSZ), 32, 0, stream>>>(
            Wh, e1, e2, e2mx, maskw, nsb[l]);
        nsin = nsb[l];
    }

    out_kernel<<<NB, TB, 0, stream>>>(nsb[0], nsb[1], nsb[2], feat, Wout, bout, out);
}